// NewAttentionGain_45397804319310
// MI455X (gfx1250) — compile-verified
//
#include <hip/hip_runtime.h>
#include <math.h>

// Problem constants
#define DIMX 512
#define DIMZ 256
#define DM   256   // d_model == en
#define NH   8
#define HD   32
#define BS   32
#define STOT 768   // DIMX + DIMZ

typedef float v2f __attribute__((ext_vector_type(2)));
typedef float v8f __attribute__((ext_vector_type(8)));

// ---------------------------------------------------------------------------
// Kernel 1: ensemble means  mu_state[b,x] = mean_e state[b,e,x], same for obs
// ---------------------------------------------------------------------------
__global__ __launch_bounds__(512) void means_kernel(
    const float* __restrict__ state, const float* __restrict__ obs,
    float* __restrict__ mu_state, float* __restrict__ mu_obs) {
  int b = blockIdx.x;
  int x = threadIdx.x;                       // 512 threads
  float s = 0.f;
  for (int e = 0; e < DM; ++e) s += state[((size_t)b * DM + e) * DIMX + x];
  mu_state[b * DIMX + x] = s * (1.0f / DM);
  if (x < DIMZ) {
    float so = 0.f;
    for (int e = 0; e < DM; ++e) so += obs[((size_t)b * DM + e) * DIMZ + x];
    mu_obs[b * DIMZ + x] = so * (1.0f / DM);
  }
}

// ---------------------------------------------------------------------------
// Kernel 2: sinusoidal positional encoding table pe[32][256]
// ---------------------------------------------------------------------------
__global__ void pe_kernel(float* __restrict__ pe) {
  int idx = blockIdx.x * blockDim.x + threadIdx.x;  // BS*DM = 8192
  int p = idx / DM, d = idx % DM;
  int i2 = d >> 1;
  // div[i] = exp(2i * (-ln(10000)/d_model))
  float freq = expf((float)(2 * i2) * (-9.210340371976184f / (float)DM));
  float v = (float)p * freq;
  pe[idx] = (d & 1) ? cosf(v) : sinf(v);
}

// ---------------------------------------------------------------------------
// Kernel 3: peWq[b,j] = pe[b]·Wq[j,:] + bq[j];  peWk likewise (+bk);
//           wkRowSum[j] = sum_e Wk[j,e]
// ---------------------------------------------------------------------------
__global__ void pew_kernel(const float* __restrict__ pe,
                           const float* __restrict__ Wq, const float* __restrict__ bq,
                           const float* __restrict__ Wk, const float* __restrict__ bk,
                           float* __restrict__ peWq, float* __restrict__ peWk,
                           float* __restrict__ wkRowSum) {
  int idx = blockIdx.x * blockDim.x + threadIdx.x;  // 8192
  int b = idx / DM, j = idx % DM;
  const float* per = pe + b * DM;
  const float* wqr = Wq + j * DM;
  const float* wkr = Wk + j * DM;
  float sq = bq[j], sk = bk[j], rs = 0.f;
  for (int d = 0; d < DM; ++d) {
    float pv = per[d];
    sq += pv * wqr[d];
    sk += pv * wkr[d];
    rs += wkr[d];
  }
  peWq[idx] = sq;
  peWk[idx] = sk;
  if (b == 0) wkRowSum[j] = rs;
}

// ---------------------------------------------------------------------------
// Kernel 4: QW[l,j] = Qp[l,:]·Wq[j,:]   (Qp @ Wq^T, batch-independent)
// ---------------------------------------------------------------------------
__global__ __launch_bounds__(256) void qw_kernel(const float* __restrict__ Qp,
                                                 const float* __restrict__ Wq,
                                                 float* __restrict__ QW) {
  __shared__ float qrow[DM];
  int l = blockIdx.x, j = threadIdx.x;
  qrow[j] = Qp[l * DM + j];
  __syncthreads();
  const float* wqr = Wq + j * DM;
  float s = 0.f;
  for (int d = 0; d < DM; ++d) s += qrow[d] * wqr[d];
  QW[l * DM + j] = s;
}

// ---------------------------------------------------------------------------
// Kernel 5: fused attention-gain kernel.
// grid = (32 row-tiles, 32 batches), block = 256 (8 wave32).
// Rows l < 256: mask admits one entry -> weight 1 exactly (copy path).
// Rows l >= 256: state & obs K-projections computed in ONE fused K-loop so
// every Wk B-fragment is loaded once and feeds 4 WMMAs (k1/k2 x 2 j-tiles).
// Then per-head 2-way softmax, scalar blend, dense atten row write.
// ---------------------------------------------------------------------------
__global__ __launch_bounds__(256) void fused_kernel(
    const float* __restrict__ state, const float* __restrict__ obs,
    const float* __restrict__ Wk,
    const float* __restrict__ mu_state, const float* __restrict__ mu_obs,
    const float* __restrict__ peWq, const float* __restrict__ peWk,
    const float* __restrict__ wkRowSum, const float* __restrict__ QW,
    float* __restrict__ out_new_state, float* __restrict__ out_atten) {
  const int t = blockIdx.x;     // row tile
  const int b = blockIdx.y;     // batch
  const int x0 = t * 16;
  const int tid = threadIdx.x;

  // --------------------- trivial rows (l < 256): weight == 1 ----------------
  if (x0 < (DIMX - DIMZ)) {
    for (int idx = tid; idx < 16 * STOT; idx += 256) {
      int m = idx / STOT, s = idx % STOT;
      int l = x0 + m;
      out_atten[((size_t)b * DIMX + l) * STOT + s] = (s == l) ? 1.0f : 0.0f;
    }
    int m = tid & 15;
    int l = x0 + m;
    for (int e = tid >> 4; e < DM; e += 16) {
      size_t off = ((size_t)b * DM + e) * DIMX + l;
      out_new_state[off] = state[off];
    }
    return;
  }

  // --------------------- full path (l >= 256) -------------------------------
  __shared__ float a1_lds[DM * 16];    // state A tile, [e][m] layout
  __shared__ float a2_lds[DM * 16];    // obs   A tile, [e][m] layout
  __shared__ float k1_lds[16 * DM];    // state-branch K rows, [m][j]
  __shared__ float k2_lds[16 * DM];    // obs-branch   K rows, [m][j]
  __shared__ float q_lds[16 * DM];     // q rows,              [m][j]
  __shared__ float red[2 * 16 * NH];   // per-(m,h) softmax weights
  __shared__ float w1_sh[16], w2_sh[16];

  const int z0 = x0 - (DIMX - DIMZ);   // obs row base
  const int lane = tid & 31;
  const int wave = tid >> 5;
  const int hi = lane >> 4;            // which K-half of the fragment
  const int n  = lane & 15;            // M for A-frag, N for B/C-frags

  // stage both A tiles and the q tile
  {
    int m = tid & 15;
    for (int e = tid >> 4; e < DM; e += 16) {
      a1_lds[e * 16 + m] = state[((size_t)b * DM + e) * DIMX + x0 + m];
      a2_lds[e * 16 + m] = obs[((size_t)b * DM + e) * DIMZ + z0 + m];
    }
  }
  {
    int m = tid >> 4;
    int jb = (tid & 15) * 16;
    for (int j = jb; j < jb + 16; ++j)
      q_lds[m * DM + j] = QW[(x0 + m) * DM + j] + peWq[b * DM + j];
  }
  __syncthreads();

  const int j0a = (wave * 2) * 16;
  const int j0b = (wave * 2 + 1) * 16;

  // ---- fused GEMMs: K1 = state^T @ Wk^T, K2 = obs^T @ Wk^T ----
  // Each B fragment (Wk tile) is loaded once and used by both branches.
  v8f acc1a = {0, 0, 0, 0, 0, 0, 0, 0};   // k1, j-tile a
  v8f acc1b = {0, 0, 0, 0, 0, 0, 0, 0};   // k1, j-tile b
  v8f acc2a = {0, 0, 0, 0, 0, 0, 0, 0};   // k2, j-tile a
  v8f acc2b = {0, 0, 0, 0, 0, 0, 0, 0};   // k2, j-tile b
  for (int e0 = 0; e0 < DM; e0 += 4) {
    v2f a1, a2;
    a1.x = a1_lds[(e0 + 2 * hi) * 16 + n];
    a1.y = a1_lds[(e0 + 2 * hi + 1) * 16 + n];
    a2.x = a2_lds[(e0 + 2 * hi) * 16 + n];
    a2.y = a2_lds[(e0 + 2 * hi + 1) * 16 + n];
    v2f b0 = *(const v2f*)&Wk[(size_t)(j0a + n) * DM + e0 + 2 * hi];
    v2f b1 = *(const v2f*)&Wk[(size_t)(j0b + n) * DM + e0 + 2 * hi];
    acc1a = __builtin_amdgcn_wmma_f32_16x16x4_f32(false, a1, false, b0,
                                                  (short)0, acc1a, false, false);
    acc1b = __builtin_amdgcn_wmma_f32_16x16x4_f32(false, a1, false, b1,
                                                  (short)0, acc1b, false, false);
    acc2a = __builtin_amdgcn_wmma_f32_16x16x4_f32(false, a2, false, b0,
                                                  (short)0, acc2a, false, false);
    acc2b = __builtin_amdgcn_wmma_f32_16x16x4_f32(false, a2, false, b1,
                                                  (short)0, acc2b, false, false);
  }

  // write with mean-centering / PE / bias corrections
  for (int r = 0; r < 8; ++r) {
    int m = r + 8 * hi;
    float rs_a = wkRowSum[j0a + n], rs_b = wkRowSum[j0b + n];
    float pk_a = peWk[b * DM + j0a + n], pk_b = peWk[b * DM + j0b + n];
    float mu1 = mu_state[b * DIMX + x0 + m];
    float mu2 = mu_obs[b * DIMZ + z0 + m];
    k1_lds[m * DM + j0a + n] = acc1a[r] - mu1 * rs_a + pk_a;
    k1_lds[m * DM + j0b + n] = acc1b[r] - mu1 * rs_b + pk_b;
    k2_lds[m * DM + j0a + n] = acc2a[r] - mu2 * rs_a + pk_a;
    k2_lds[m * DM + j0b + n] = acc2b[r] - mu2 * rs_b + pk_b;
  }
  __syncthreads();

  // ---- per-head scores + 2-way softmax ----
  if (tid < 16 * NH) {
    int m = tid >> 3, h = tid & 7;
    const float* qr = &q_lds[m * DM + h * HD];
    const float* k1 = &k1_lds[m * DM + h * HD];
    const float* k2 = &k2_lds[m * DM + h * HD];
    float s1 = 0.f, s2 = 0.f;
    for (int d = 0; d < HD; ++d) {
      s1 += qr[d] * k1[d];
      s2 += qr[d] * k2[d];
    }
    const float scale = 0.17677669529663687f;  // 1/sqrt(32)
    s1 *= scale;
    s2 *= scale;
    red[m * NH + h]       = 1.0f / (1.0f + expf(s2 - s1));  // w1 per head
    red[128 + m * NH + h] = 1.0f / (1.0f + expf(s1 - s2));  // w2 per head
  }
  __syncthreads();
  if (tid < 16) {
    float a1 = 0.f, a2 = 0.f;
    for (int h = 0; h < NH; ++h) {
      a1 += red[tid * NH + h];
      a2 += red[128 + tid * NH + h];
    }
    w1_sh[tid] = a1 * (1.0f / NH);
    w2_sh[tid] = a2 * (1.0f / NH);
  }
  __syncthreads();

  // ---- write dense atten rows (zeros + two entries) ----
  for (int idx = tid; idx < 16 * STOT; idx += 256) {
    int m = idx / STOT, s = idx % STOT;
    int l = x0 + m;
    float v = 0.f;
    if (s == l) v = w1_sh[m];
    else if (s == l + DIMZ) v = w2_sh[m];   // 512 + (l-256)
    out_atten[((size_t)b * DIMX + l) * STOT + s] = v;
  }

  // ---- blended new_state columns ----
  {
    int m = tid & 15;
    int l = x0 + m;
    float w1 = w1_sh[m], w2 = w2_sh[m];
    for (int e = tid >> 4; e < DM; e += 16) {
      float v = w1 * state[((size_t)b * DM + e) * DIMX + l] +
                w2 * obs[((size_t)b * DM + e) * DIMZ + (z0 + m)];
      out_new_state[((size_t)b * DM + e) * DIMX + l] = v;
    }
  }
}

// ---------------------------------------------------------------------------
extern "C" void kernel_launch(void* const* d_in, const int* in_sizes, int n_in,
                              void* d_out, int out_size, void* d_ws, size_t ws_size,
                              hipStream_t stream) {
  const float* state = (const float*)d_in[0];  // [32,256,512]
  const float* obs   = (const float*)d_in[1];  // [32,256,256]
  const float* Qp    = (const float*)d_in[2];  // [512,256]
  const float* Wq    = (const float*)d_in[3];  // [256,256]
  const float* bq    = (const float*)d_in[4];  // [256]
  const float* Wk    = (const float*)d_in[5];  // [256,256]
  const float* bk    = (const float*)d_in[6];  // [256]

  float* new_state = (float*)d_out;                          // [32,256,512]
  float* atten     = (float*)d_out + (size_t)BS * DM * DIMX; // [32,512,768]

  // workspace layout (floats)
  float* w = (float*)d_ws;
  float* mu_state = w;                     // 32*512
  float* mu_obs   = mu_state + BS * DIMX;  // 32*256
  float* pe       = mu_obs + BS * DIMZ;    // 32*256
  float* peWq     = pe + BS * DM;          // 32*256
  float* peWk     = peWq + BS * DM;        // 32*256
  float* wkRowSum = peWk + BS * DM;        // 256
  float* QW       = wkRowSum + DM;         // 512*256

  means_kernel<<<BS, DIMX, 0, stream>>>(state, obs, mu_state, mu_obs);
  pe_kernel<<<BS, DM, 0, stream>>>(pe);
  pew_kernel<<<BS, DM, 0, stream>>>(pe, Wq, bq, Wk, bk, peWq, peWk, wkRowSum);
  qw_kernel<<<DIMX, DM, 0, stream>>>(Qp, Wq, QW);
  fused_kernel<<<dim3(DIMX / 16, BS), 256, 0, stream>>>(
      state, obs, Wk, mu_state, mu_obs, peWq, peWk, wkRowSum, QW,
      new_state, atten);
}